// ACWLoss_81673098101676
// MI455X (gfx1250) — compile-verified
//
#include <hip/hip_runtime.h>
#include <hip/hip_bf16.h>
#include <math.h>
#include <stdint.h>

// Problem constants (fixed by the reference)
#define CH   19
#define HW_  (512 * 512)
#define ALPHA_ 1.2f
#define BETA_  0.8f
#define EPS_   1e-7f

#define AS1 __attribute__((address_space(1)))
#define AS3 __attribute__((address_space(3)))

typedef int v4i_ __attribute__((vector_size(16)));    // matches async builtin's int4 pointee
typedef unsigned uv4_ __attribute__((vector_size(16)));  // TDM D# group0
typedef int iv4_ __attribute__((vector_size(16)));    // TDM D# groups 2/3
typedef int iv8_ __attribute__((vector_size(32)));    // TDM D# group1

// CDNA5 async global->LDS path (ASYNCcnt), guarded so the file always compiles.
#if defined(__gfx1250__) && __has_builtin(__builtin_amdgcn_global_load_async_to_lds_b128) && \
    __has_builtin(__builtin_amdgcn_s_wait_asynccnt)
#define HAVE_ASYNC_LDS 1
#endif

// CDNA5 Tensor Data Mover path (TENSORcnt), guarded likewise.
#if defined(__gfx1250__) && __has_builtin(__builtin_amdgcn_tensor_load_to_lds) && \
    __has_builtin(__builtin_amdgcn_s_wait_tensorcnt)
#define HAVE_TDM 1
#endif

// ---------------------------------------------------------------------------
// Pass 1: per-pixel softmax -> p_true, fused with the hi-16 radix histogram.
// Each thread owns 4 adjacent pixels so every channel-plane read is a 16B
// global_load_b128; all 19x4 logits live in VGPRs.  Dynamic indexing by the
// target class is replaced with a select inside the unrolled channel loop.
// ---------------------------------------------------------------------------
__global__ __launch_bounds__(256) void ptrue_kernel(const float* __restrict__ x,
                                                    const int* __restrict__ tgt,
                                                    float* __restrict__ ptrue,
                                                    unsigned* __restrict__ hist,
                                                    int nquad) {
  int q = blockIdx.x * blockDim.x + threadIdx.x;
  if (q >= nquad) return;
  long long n0 = (long long)q * 4;
  int b = (int)(n0 / HW_);
  long long hw = n0 - (long long)b * HW_;   // quads never straddle a batch (HW%4==0)
  const float* base = x + (long long)b * CH * HW_ + hw;

  // gfx1250 prefetch path (lowers to global_prefetch_b8)
  __builtin_prefetch(tgt + n0, 0, 0);

  float a[CH][4];
#pragma unroll
  for (int c = 0; c < CH; ++c) {
    float4 v = *reinterpret_cast<const float4*>(base + (long long)c * HW_);
    a[c][0] = v.x; a[c][1] = v.y; a[c][2] = v.z; a[c][3] = v.w;
  }

  int4 t4 = *reinterpret_cast<const int4*>(tgt + n0);
  int tj[4] = {t4.x, t4.y, t4.z, t4.w};

  float out[4];
#pragma unroll
  for (int j = 0; j < 4; ++j) {
    float m = a[0][j];
#pragma unroll
    for (int c = 1; c < CH; ++c) m = fmaxf(m, a[c][j]);
    float s = 0.0f, vt = 0.0f;
#pragma unroll
    for (int c = 0; c < CH; ++c) {
      float e = expf(a[c][j] - m);
      s += e;
      vt = (c == tj[j]) ? e : vt;   // select instead of dynamic reg index
    }
    out[j] = vt / s;
  }
  *reinterpret_cast<float4*>(ptrue + n0) = make_float4(out[0], out[1], out[2], out[3]);

  // Fused hi-16 histogram (saves a full re-read of p_true)
#pragma unroll
  for (int j = 0; j < 4; ++j) {
    unsigned u = __float_as_uint(out[j]);
    atomicAdd(&hist[u >> 16], 1u);
  }
}

// ---------------------------------------------------------------------------
// Exact quantile via 2-level radix select on float bit patterns.
// p_true in (0,1] -> all positive -> uint32 bit pattern order == float order.
// ---------------------------------------------------------------------------
__global__ void zero_u32(unsigned* p, int n) {
  int i = blockIdx.x * blockDim.x + threadIdx.x;
  if (i < n) p[i] = 0u;
}

// Find hi-16 bins holding ranks r0 and r1 in one serial scan.
// out: [0]=bin0 [1]=rem0 [2]=bin1 [3]=rem1
__global__ void find_hi2(const unsigned* __restrict__ hist, unsigned r0, unsigned r1,
                         unsigned* __restrict__ out) {
  if (blockIdx.x != 0 || threadIdx.x != 0) return;
  unsigned cum = 0, b0 = 65535u, rem0 = 0, b1 = 65535u, rem1 = 0;
  bool f0 = false, f1 = false;
  for (unsigned b = 0; b < 65536u; ++b) {
    unsigned c = hist[b];
    if (!f0 && cum + c > r0) { b0 = b; rem0 = r0 - cum; f0 = true; }
    if (!f1 && cum + c > r1) { b1 = b; rem1 = r1 - cum; f1 = true; }
    cum += c;
    if (f0 && f1) break;
  }
  out[0] = b0; out[1] = rem0; out[2] = b1; out[3] = rem1;
}

// ---------------------------------------------------------------------------
// lo-16 histogram pass: each block consumes one contiguous 8KB run of p_true,
// DMA'd into LDS by the Tensor Data Mover (tensor_load_to_lds, TENSORcnt).
// D# layout per CDNA5 ISA: group0 = count/lds/global/
// type, group1 = data_size + tensor/tile dims + strides.  2D descriptor with
// tile_dim1 = 1; groups 2/3 zero.
// ---------------------------------------------------------------------------
#define TILE_F 2048   // floats per TDM tile (8KB LDS)

__global__ __launch_bounds__(256) void hist_lo(const float* __restrict__ pt, int N,
                                               const unsigned* __restrict__ binrem,
                                               unsigned* __restrict__ hist) {
  unsigned hi = binrem[0];
#if defined(HAVE_TDM)
  __shared__ float tile[TILE_F];
  unsigned lds_off = (unsigned)(uintptr_t)(AS3 float*)tile;  // 32-bit LDS byte offset
  for (long long base = (long long)blockIdx.x * TILE_F; base < N;
       base += (long long)gridDim.x * TILE_F) {
    long long rem = N - base;
    int valid = rem < TILE_F ? (int)rem : TILE_F;
    if (threadIdx.x == 0) {
      unsigned long long ga = (unsigned long long)(uintptr_t)(pt + base);
      uv4_ g0 = {
        1u,                                              // count=1, user mode
        lds_off,                                         // lds_addr
        (unsigned)(ga & 0xFFFFFFFFull),                  // global_addr[31:0]
        (unsigned)((ga >> 32) & 0x01FFFFFFull) | 0x80000000u  // ga[56:32] | type=2
      };
      iv8_ g1 = {
        (int)(2u << 16),                                 // data_size = 4B
        (int)(((unsigned)valid & 0xFFFFu) << 16),        // tensor_dim0[15:0]
        (int)((((unsigned)valid >> 16) & 0xFFFFu) | (1u << 16)), // td0[31:16] | tensor_dim1=1
        (int)(((unsigned)TILE_F) << 16),                 // tile_dim0
        1,                                               // tile_dim1=1, tile_dim2=0
        TILE_F,                                          // tensor_dim0_stride[31:0]
        0, 0                                             // stride hi, dim1_stride
      };
      iv4_ gz = {0, 0, 0, 0};
#if __clang_major__ >= 23
      iv8_ g4 = {0, 0, 0, 0, 0, 0, 0, 0};
      __builtin_amdgcn_tensor_load_to_lds(g0, g1, gz, gz, g4, 0);
#else
      __builtin_amdgcn_tensor_load_to_lds(g0, g1, gz, gz, 0);
#endif
      __builtin_amdgcn_s_wait_tensorcnt(0);
    }
    __syncthreads();
    for (int j = threadIdx.x; j < valid; j += 256) {
      unsigned u = __float_as_uint(tile[j]);
      if ((u >> 16) == hi) atomicAdd(&hist[u & 0xFFFFu], 1u);
    }
    __syncthreads();   // protect tile before next DMA overwrites it
  }
#else
  for (int i = blockIdx.x * blockDim.x + threadIdx.x; i < N;
       i += gridDim.x * blockDim.x) {
    unsigned u = __float_as_uint(pt[i]);
    if ((u >> 16) == hi) atomicAdd(&hist[u & 0xFFFFu], 1u);
  }
#endif
}

__global__ void find_lo(const unsigned* __restrict__ hist,
                        const unsigned* __restrict__ binrem,
                        float* __restrict__ dst) {
  if (blockIdx.x != 0 || threadIdx.x != 0) return;
  unsigned hi = binrem[0], rem = binrem[1];
  unsigned cum = 0, l = 65535u;
  for (unsigned i = 0; i < 65536u; ++i) {
    unsigned c = hist[i];
    if (cum + c > rem) { l = i; break; }
    cum += c;
  }
  dst[0] = __uint_as_float((hi << 16) | l);
}

// th = v_k + frac*(v_{k+1} - v_k)   (jnp.quantile 'linear' interpolation)
__global__ void thresh_kernel(const float* __restrict__ vk, float frac,
                              float* __restrict__ th) {
  if (blockIdx.x != 0 || threadIdx.x != 0) return;
  th[0] = vk[0] + frac * (vk[1] - vk[0]);
}

// ---------------------------------------------------------------------------
// Masked-mean reduction, streaming p_true through LDS via the CDNA5 async
// DMA path (GLOBAL_LOAD_ASYNC_TO_LDS_B128 + s_wait_asynccnt).  Each lane DMAs
// 16B into its private LDS slot and reads back only its own slot, so the
// per-wave ASYNCcnt wait is sufficient (no workgroup barrier needed).
// Deterministic: fixed grid, per-block LDS tree reduction into a partials
// array, then a fixed-order final accumulation (no float global atomics).
// ---------------------------------------------------------------------------
#define RED_BLOCK 256
#define RED_CHUNK (RED_BLOCK * 4)   // floats per block-iteration

__global__ __launch_bounds__(RED_BLOCK) void reduce_masks(const float* __restrict__ pt, int N,
                                                          const float* __restrict__ thp,
                                                          float4* __restrict__ partials) {
  __shared__ float tile[RED_CHUNK];
  float th = *thp;
  float sh = 0.f, ch = 0.f, sl = 0.f, cl = 0.f;
  const int tid = threadIdx.x;

  for (long long base = (long long)blockIdx.x * RED_CHUNK; base < N;
       base += (long long)gridDim.x * RED_CHUNK) {
    if (base + RED_CHUNK <= N) {
      float4 v;
#if defined(HAVE_ASYNC_LDS)
      // async DMA: global -> LDS (tracked by ASYNCcnt, not LOADcnt)
      __builtin_amdgcn_global_load_async_to_lds_b128(
          (AS1 v4i_*)(pt + base + (long long)tid * 4),
          (AS3 v4i_*)(tile + tid * 4), 0, 0);
      __builtin_amdgcn_s_wait_asynccnt(0);
      v = *reinterpret_cast<float4*>(&tile[tid * 4]);
#else
      v = *reinterpret_cast<const float4*>(pt + base + (long long)tid * 4);
#endif
      float e[4] = {v.x, v.y, v.z, v.w};
#pragma unroll
      for (int j = 0; j < 4; ++j) {
        float p = e[j];
        float nl = -logf(p + EPS_);
        if (p > th) { sh += nl; ch += 1.f; }
        else        { sl += nl; cl += 1.f; }
      }
    } else {
      for (long long i = base + tid; i < N; i += RED_BLOCK) {
        float p = pt[i];
        float nl = -logf(p + EPS_);
        if (p > th) { sh += nl; ch += 1.f; }
        else        { sl += nl; cl += 1.f; }
      }
    }
  }

  __shared__ float s0[RED_BLOCK], s1[RED_BLOCK], s2[RED_BLOCK], s3[RED_BLOCK];
  s0[tid] = sh; s1[tid] = ch; s2[tid] = sl; s3[tid] = cl;
  __syncthreads();
  for (int off = RED_BLOCK / 2; off > 0; off >>= 1) {
    if (tid < off) {
      s0[tid] += s0[tid + off];
      s1[tid] += s1[tid + off];
      s2[tid] += s2[tid + off];
      s3[tid] += s3[tid + off];
    }
    __syncthreads();
  }
  if (tid == 0) partials[blockIdx.x] = make_float4(s0[0], s1[0], s2[0], s3[0]);
}

__global__ void finalize(const float4* __restrict__ partials, int nb,
                         float* __restrict__ out) {
  if (blockIdx.x != 0 || threadIdx.x != 0) return;
  double sh = 0, ch = 0, sl = 0, cl = 0;
  for (int i = 0; i < nb; ++i) {
    float4 p = partials[i];
    sh += p.x; ch += p.y; sl += p.z; cl += p.w;
  }
  double la = (ch > 0.0) ? sh / fmax(ch, 1.0) : 0.0;
  double lb = (cl > 0.0) ? sl / fmax(cl, 1.0) : 0.0;
  out[0] = (float)(la * (double)ALPHA_ + lb * (double)BETA_);
}

// ---------------------------------------------------------------------------
// Host-side orchestration (single stream; graph-capture safe).
// Workspace layout: [p_true: N*4B][hist: 256KB][scalars: 128B][partials: 16KB]
// ---------------------------------------------------------------------------
extern "C" void kernel_launch(void* const* d_in, const int* in_sizes, int n_in,
                              void* d_out, int out_size, void* d_ws, size_t ws_size,
                              hipStream_t stream) {
  const float* x = (const float*)d_in[0];
  const int* tgt = (const int*)d_in[1];
  float* out = (float*)d_out;

  const int N = in_sizes[1];         // 2,097,152 pixels
  const int nquad = N / 4;

  char* ws = (char*)d_ws;
  float* ptrue = (float*)ws;
  unsigned* hist = (unsigned*)(ws + (size_t)N * 4);
  unsigned* su   = (unsigned*)((char*)hist + 65536u * 4);   // bin/rem scalars
  float* sf      = (float*)((char*)su + 64);                // vk, vk1, th
  float4* partials = (float4*)((char*)sf + 64);

  double idx = 0.7 * (double)(N - 1);
  unsigned k = (unsigned)idx;
  float frac = (float)(idx - (double)k);

  // 1. softmax -> p_true fused with hi-16 histogram (the only heavy pass)
  zero_u32<<<256, 256, 0, stream>>>(hist, 65536);
  ptrue_kernel<<<(nquad + 255) / 256, 256, 0, stream>>>(x, tgt, ptrue, hist, nquad);
  find_hi2<<<1, 32, 0, stream>>>(hist, k, k + 1, su);

  // 2. lo-16 refinement for exact order stats at ranks k and k+1 (TDM-staged)
  zero_u32<<<256, 256, 0, stream>>>(hist, 65536);
  hist_lo<<<1024, 256, 0, stream>>>(ptrue, N, su + 0, hist);
  find_lo<<<1, 32, 0, stream>>>(hist, su + 0, sf + 0);

  zero_u32<<<256, 256, 0, stream>>>(hist, 65536);
  hist_lo<<<1024, 256, 0, stream>>>(ptrue, N, su + 2, hist);
  find_lo<<<1, 32, 0, stream>>>(hist, su + 2, sf + 1);

  thresh_kernel<<<1, 32, 0, stream>>>(sf, frac, sf + 2);

  // 3. masked means + combine (async-LDS staged stream)
  reduce_masks<<<1024, RED_BLOCK, 0, stream>>>(ptrue, N, sf + 2, partials);
  finalize<<<1, 32, 0, stream>>>(partials, 1024, out);
}